// CoAttention_18339510354097
// MI455X (gfx1250) — compile-verified
//
#include <hip/hip_runtime.h>
#include <hip/hip_bf16.h>

typedef __attribute__((ext_vector_type(2))) float v2f;
typedef __attribute__((ext_vector_type(8))) float v8f;

#define HW  196
#define TT  64
#define NB  16
#define MD  2048
#define HD  768
#define AD  512

// ---------------------------------------------------------------------------
// Generic fp32 WMMA GEMM:  C = A(MxK,row-major) * B(KxN,row-major)
//                             [+ bias[col]] [* mul[row*ldc+col]]
// block = 128 threads (4 waves). Block tile 64(M) x 64(N); wave tile 16x64.
// K consumed in chunks of 32 (8 x WMMA_16x16x4_f32).
// B chunks stream into a double-buffered LDS tile via CDNA5 async-to-LDS
// (GLOBAL_LOAD_ASYNC_TO_LDS_B128, ASYNCcnt pipelined one chunk ahead).
// Requires: M%64==0 per grid.y, N%64==0 per grid.x, K%4==0.
// blockIdx.z = batch index (strides sA/sB/sC/sMul in elements).
// ---------------------------------------------------------------------------
__global__ __launch_bounds__(128) void wmma_gemm_f32(
    const float* __restrict__ A, const float* __restrict__ B,
    const float* __restrict__ bias, const float* __restrict__ mul,
    float* __restrict__ C,
    int K, int lda, int ldb, int ldc,
    long sA, long sB, long sC, long sMul)
{
  __shared__ alignas(16) float Bs[2][32][64];   // 16 KB double buffer

  const int tid  = threadIdx.x;
  const int w    = tid >> 5;        // wave id 0..3
  const int lane = tid & 31;
  const int l15  = lane & 15;
  const int kh   = lane >> 4;       // 0: K pair {0,1}, 1: K pair {2,3}
  const int row0 = blockIdx.y * 64;
  const int col0 = blockIdx.x * 64;
  const int z    = blockIdx.z;

  A += (long)z * sA;
  B += (long)z * sB;
  C += (long)z * sC;
  const float* mulp = mul ? mul + (long)z * sMul : nullptr;

  // This lane's A row, pre-offset by its K-pair (ISA 16x4 f32 A layout).
  const float* arow = A + (size_t)(row0 + w * 16 + l15) * lda + 2 * kh;

  v8f zero = {0.f, 0.f, 0.f, 0.f, 0.f, 0.f, 0.f, 0.f};
  v8f acc[4];
#pragma unroll
  for (int t = 0; t < 4; ++t) acc[t] = zero;

  // Async-stage one 32x64 B chunk into LDS buffer `buf`.
  // Always issues exactly 4 wave-level async ops (rows clamped, not
  // predicated) so the per-wave ASYNCcnt bookkeeping stays uniform.
  auto issueB = [&](int k0, int buf) {
#pragma unroll
    for (int i = 0; i < 4; ++i) {
      const int id = tid + 128 * i;
      const int kk = id >> 4;              // 0..31
      const int c4 = (id & 15) << 2;       // 0,4,..,60
      int row = k0 + kk;
      if (row > K - 1) row = K - 1;        // clamp: stay in-bounds on tail
      const unsigned voff =
          ((unsigned)row * (unsigned)ldb + (unsigned)(col0 + c4)) * 4u;
      const unsigned lds = (unsigned)(size_t)&Bs[buf][kk][c4];
      asm volatile("global_load_async_to_lds_b128 %0, %1, %2 offset:0"
                   :: "v"(lds), "v"(voff), "s"(B)
                   : "memory");
    }
  };

  const int nch = (K + 31) >> 5;
  issueB(0, 0);

  for (int i = 0; i < nch; ++i) {
    const int cur = i & 1;
    const int k0  = i << 5;
    int kc = K - k0;
    if (kc > 32) kc = 32;

    if (i + 1 < nch) {
      issueB((i + 1) << 5, cur ^ 1);                     // prefetch next chunk
      asm volatile("s_wait_asynccnt 0x4" ::: "memory");  // only the 4 new remain
    } else {
      asm volatile("s_wait_asynccnt 0x0" ::: "memory");
    }
    __syncthreads();                                     // chunk i visible to all

    const float (*Bc)[64] = Bs[cur];

    auto step = [&](int j) {
      const int kb = j * 4 + 2 * kh;
      v2f a;
      a.x = arow[k0 + j * 4];
      a.y = arow[k0 + j * 4 + 1];
#pragma unroll
      for (int t = 0; t < 4; ++t) {
        v2f b;
        b.x = Bc[kb][t * 16 + l15];
        b.y = Bc[kb + 1][t * 16 + l15];
        acc[t] = __builtin_amdgcn_wmma_f32_16x16x4_f32(
            false, a, false, b, (short)0, acc[t], false, false);
      }
    };

    if (kc == 32) {
#pragma unroll
      for (int j = 0; j < 8; ++j) step(j);
    } else {
      for (int j = 0; j < (kc >> 2); ++j) step(j);
    }
    __syncthreads();   // done reading Bs[cur] before it is refilled
  }

  // Epilogue: C/D layout — VGPR r: row = r + 8*kh, col = l15.
#pragma unroll
  for (int t = 0; t < 4; ++t) {
    const int col = col0 + t * 16 + l15;
    const float bb = bias ? bias[col] : 0.f;
#pragma unroll
    for (int r = 0; r < 8; ++r) {
      const int row = row0 + w * 16 + r + 8 * kh;
      const size_t off = (size_t)row * ldc + col;
      float v = acc[t][r] + bb;
      if (mulp) v *= mulp[off];
      C[off] = v;
    }
  }
}

// ---------------------------------------------------------------------------
// scores[b,t,s] = sum_a relu(ml[b,s,a] + hl[b,t,a]) * Wr[a] + br, then
// softmax over s (196 pixels). One 256-thread block per (b,t) row.
// ---------------------------------------------------------------------------
__global__ __launch_bounds__(256) void attn_scores_softmax(
    const float* __restrict__ ml,   // (B*HW, AD)
    const float* __restrict__ hl,   // (B*TT, AD)
    const float* __restrict__ Wr,   // (AD)
    const float* __restrict__ br,   // scalar
    float* __restrict__ softmap)    // (B*TT, HW)
{
  __shared__ alignas(16) float hsh[AD];
  __shared__ alignas(16) float wsh[AD];
  __shared__ float red[256];

  const int tid = threadIdx.x;
  const int bt  = blockIdx.x;     // b*64 + t
  const int b   = bt >> 6;

  for (int i = tid; i < AD; i += 256) {
    hsh[i] = hl[(size_t)bt * AD + i];
    wsh[i] = Wr[i];
  }
  __syncthreads();

  float sc = -3.4e38f;
  if (tid < HW) {
    const float* mp = ml + (size_t)(b * HW + tid) * AD;
    float acc = 0.f;
#pragma unroll 4
    for (int a = 0; a < AD; a += 4) {
      const float4 m4 = *(const float4*)(mp + a);
      acc += fmaxf(m4.x + hsh[a + 0], 0.f) * wsh[a + 0];
      acc += fmaxf(m4.y + hsh[a + 1], 0.f) * wsh[a + 1];
      acc += fmaxf(m4.z + hsh[a + 2], 0.f) * wsh[a + 2];
      acc += fmaxf(m4.w + hsh[a + 3], 0.f) * wsh[a + 3];
    }
    sc = acc + br[0];
  }

  // max over s
  red[tid] = sc;
  __syncthreads();
  for (int s = 128; s > 0; s >>= 1) {
    if (tid < s) red[tid] = fmaxf(red[tid], red[tid + s]);
    __syncthreads();
  }
  const float mx = red[0];
  __syncthreads();

  const float e = (tid < HW) ? __expf(sc - mx) : 0.f;

  // sum over s
  red[tid] = e;
  __syncthreads();
  for (int s = 128; s > 0; s >>= 1) {
    if (tid < s) red[tid] += red[tid + s];
    __syncthreads();
  }
  const float inv = 1.f / red[0];

  if (tid < HW) softmap[(size_t)bt * HW + tid] = e * inv;
}

// ---------------------------------------------------------------------------
extern "C" void kernel_launch(void* const* d_in, const int* in_sizes, int n_in,
                              void* d_out, int out_size, void* d_ws, size_t ws_size,
                              hipStream_t stream) {
  const float* maps    = (const float*)d_in[0];  // (16,14,14,2048) = fm (3136,2048)
  const float* hiddens = (const float*)d_in[1];  // (16,64,768) = (1024,768)
  const float* Wm      = (const float*)d_in[2];  // (2048,512)
  const float* bm      = (const float*)d_in[3];  // (512)
  const float* Wh      = (const float*)d_in[4];  // (768,512)
  const float* bh      = (const float*)d_in[5];  // (512)
  const float* Wr      = (const float*)d_in[6];  // (512)
  const float* br      = (const float*)d_in[7];  // scalar
  const float* Wc      = (const float*)d_in[8];  // (2048,768)
  const float* bc      = (const float*)d_in[9];  // (768)

  // workspace: ml (3136x512) | hl (1024x512) | ctx (1024x2048)  = 16.9 MB
  float* ml  = (float*)d_ws;
  float* hl  = ml + (size_t)3136 * AD;
  float* ctx = hl + (size_t)1024 * AD;

  // outputs: co_att (1024x768) then softmap (1024x196)
  float* co_att  = (float*)d_out;
  float* softmap = co_att + (size_t)1024 * HD;

  // 1) ml = fm @ Wm + bm          (M=3136, N=512, K=2048)
  wmma_gemm_f32<<<dim3(AD / 64, 3136 / 64, 1), 128, 0, stream>>>(
      maps, Wm, bm, nullptr, ml, MD, MD, AD, AD, 0, 0, 0, 0);

  // 2) hl = hiddens @ Wh + bh     (M=1024, N=512, K=768)
  wmma_gemm_f32<<<dim3(AD / 64, 1024 / 64, 1), 128, 0, stream>>>(
      hiddens, Wh, bh, nullptr, hl, HD, HD, AD, AD, 0, 0, 0, 0);

  // 3) additive-attention scores + softmax over pixels -> softmap (in d_out)
  attn_scores_softmax<<<dim3(NB * TT), 256, 0, stream>>>(ml, hl, Wr, br, softmap);

  // 4) ctx[b] = softmap[b] @ fm[b]   (batched: M=64, N=2048, K=196)
  wmma_gemm_f32<<<dim3(MD / 64, 1, NB), 128, 0, stream>>>(
      softmap, maps, nullptr, nullptr, ctx, HW, HW, MD, MD,
      (long)TT * HW, (long)HW * MD, (long)TT * MD, 0);

  // 5) co_att = (ctx @ Wc + bc) * hiddens   (M=1024, N=768, K=2048)
  wmma_gemm_f32<<<dim3(HD / 64, 1024 / 64, 1), 128, 0, stream>>>(
      ctx, Wc, bc, hiddens, co_att, MD, MD, HD, HD, 0, 0, 0, (long)0);
}